// DGCF_44830868636068
// MI455X (gfx1250) — compile-verified
//
#include <hip/hip_runtime.h>
#include <hip/hip_bf16.h>
#include <stdint.h>

// Problem constants (match reference)
#define NU 50000
#define NI 50000
#define NN 100000            // NU + NI
#define DD 128               // embedding dim
#define NF 4                 // factors
#define CD 32                // chunk dim = DD/NF
#define NE 2000000           // edges
#define EPSF 1e-12f

// Flip to 0 to fall back to plain VGPR loads (no async-to-LDS staging).
#define USE_ASYNC_STAGE 1
#define EPW 4                // edges per wave in k_prop (software pipeline)

// ---------- wave32 butterfly sum (all lanes get the result) ----------
__device__ __forceinline__ float wsum32(float v) {
#pragma unroll
    for (int m = 16; m > 0; m >>= 1) v += __shfl_xor(v, m, 32);
    return v;
}

__device__ __forceinline__ void softmax4(float a0, float a1, float a2, float a3,
                                         float& s0, float& s1, float& s2, float& s3) {
    float m = fmaxf(fmaxf(a0, a1), fmaxf(a2, a3));
    s0 = __expf(a0 - m); s1 = __expf(a1 - m);
    s2 = __expf(a2 - m); s3 = __expf(a3 - m);
    float inv = 1.0f / (s0 + s1 + s2 + s3);
    s0 *= inv; s1 *= inv; s2 *= inv; s3 *= inv;
}

#if USE_ASYNC_STAGE
// Stage 512B (one embedding row) into LDS: lane l moves bytes [16l,16l+16).
// GVS addressing: mem = SADDR(base) + VADDR32. Tracked by ASYNCcnt (in-order).
__device__ __forceinline__ void async_stage_row(uint32_t ldsByte, uint32_t row,
                                                const float* __restrict__ base,
                                                int lane) {
    uint32_t l = ldsByte + (uint32_t)lane * 16u;
    uint32_t g = row * (DD * 4u) + (uint32_t)lane * 16u;
    asm volatile("global_load_async_to_lds_b128 %0, %1, %2"
                 :: "v"(l), "v"(g), "s"(base) : "memory");
}
__device__ __forceinline__ void wait_async0() {
    asm volatile("s_wait_asynccnt 0x0" ::: "memory");
}
__device__ __forceinline__ void wait_async1() {
    asm volatile("s_wait_asynccnt 0x1" ::: "memory");
}
#endif

// ---------- ego = [user;item]; accum = ego ----------
__global__ void k_init(const float* __restrict__ ue, const float* __restrict__ ie,
                       float* __restrict__ ego, float* __restrict__ acc) {
    size_t i = (size_t)blockIdx.x * blockDim.x + threadIdx.x;   // < NN*DD
    float v = (i < (size_t)NU * DD) ? ue[i] : ie[i - (size_t)NU * DD];
    ego[i] = v; acc[i] = v;
}

__global__ void k_fill1(float* __restrict__ A) {
    size_t i = (size_t)blockIdx.x * blockDim.x + threadIdx.x;   // < NF*NE
    A[i] = 1.0f;
}

__global__ void k_zero(float* __restrict__ p, long n) {
    long i = (long)blockIdx.x * blockDim.x + threadIdx.x;
    if (i < n) p[i] = 0.0f;
}

// ---------- first-iteration degrees: A==1 => scores==0.25 (no A read) ----------
__global__ void k_deg0(const int* __restrict__ hl, float* __restrict__ deg) {
    int e = blockIdx.x * 256 + threadIdx.x;
    if (e >= NE) return;
    __builtin_prefetch(&hl[e + 4096], 0, 0);       // global_prefetch_b8
    int h = hl[e];
    atomicAdd(&deg[h],          0.25f);
    atomicAdd(&deg[NN + h],     0.25f);
    atomicAdd(&deg[2 * NN + h], 0.25f);
    atomicAdd(&deg[3 * NN + h], 0.25f);
}

// ---------- generic degree pass (small-workspace fallback only) ----------
__global__ void k_deg(const float* __restrict__ A, const int* __restrict__ hl,
                      float* __restrict__ deg) {
    int e = blockIdx.x * 256 + threadIdx.x;
    if (e >= NE) return;
    __builtin_prefetch(&hl[e + 4096], 0, 0);
    __builtin_prefetch(&A[e + 4096], 0, 0);
    float s0, s1, s2, s3;
    softmax4(A[e], A[NE + e], A[2 * (size_t)NE + e], A[3 * (size_t)NE + e], s0, s1, s2, s3);
    int h = hl[e];
    atomicAdd(&deg[h],          s0);
    atomicAdd(&deg[NN + h],     s1);
    atomicAdd(&deg[2 * NN + h], s2);
    atomicAdd(&deg[3 * NN + h], s3);
}

// ---------- deg -> d_inv in place ----------
__global__ void k_dinv(float* __restrict__ d) {
    int i = blockIdx.x * 256 + threadIdx.x;
    if (i >= NF * NN) return;
    d[i] = 1.0f / sqrtf(fmaxf(d[i], EPSF));
}

// ---------- per-layer hoist: T[n][d] = tanh( ego[n][d] / max(||chunk||,eps) ) ----------
__global__ void k_egoprep(const float* __restrict__ ego, float* __restrict__ T) {
    const int lane = threadIdx.x & 31;
    const int n    = blockIdx.x * 8 + (threadIdx.x >> 5);   // grid = NN/8 exactly
#pragma unroll
    for (int f = 0; f < NF; ++f) {
        size_t idx = (size_t)n * DD + f * CD + lane;
        float v = ego[idx];
        float s = wsum32(v * v);
        float rinv = 1.0f / fmaxf(sqrtf(s), EPSF);
        T[idx] = tanhf(v * rinv);
    }
}

// ---------- propagate: x[h] += s_f * d_inv[f][t] * ego[t] ----------
// EPW edges per wave, 2-deep double-buffered async pipeline: stage edge k+1's
// 512B ego[t] row while edge k's 128 atomic adds drain. In-order ASYNCcnt
// completion => s_wait_asynccnt 1 proves the current buffer is resident.
__global__ void k_prop(const float* __restrict__ A, const int* __restrict__ hl,
                       const int* __restrict__ tl, const float* __restrict__ dinv,
                       const float* __restrict__ ego, float* __restrict__ x) {
#if USE_ASYNC_STAGE
    __shared__ float stage[8 * 2 * DD];            // 8 waves x 2 buffers x 512B
#endif
    const int lane = threadIdx.x & 31;
    const int wid  = threadIdx.x >> 5;
    const int e0   = (blockIdx.x * 8 + wid) * EPW; // grid = NE/(8*EPW) exactly
    const int f    = lane >> 3;                    // 0..3

    __builtin_prefetch(&hl[e0 + 2048], 0, 0);
    __builtin_prefetch(&tl[e0 + 2048], 0, 0);

    int tv[EPW];
#pragma unroll
    for (int k = 0; k < EPW; ++k) tv[k] = tl[e0 + k];

#if USE_ASYNC_STAGE
    const uint32_t ldsBase = (uint32_t)(uintptr_t)(&stage[wid * 2 * DD]);
    async_stage_row(ldsBase, (uint32_t)tv[0], ego, lane);   // prologue
#endif

#pragma unroll
    for (int k = 0; k < EPW; ++k) {
        const int e = e0 + k;
        const int h = hl[e];
        float s0, s1, s2, s3;
        softmax4(A[e], A[NE + e], A[2 * (size_t)NE + e], A[3 * (size_t)NE + e],
                 s0, s1, s2, s3);
        float sf = (f == 0) ? s0 : (f == 1) ? s1 : (f == 2) ? s2 : s3;
        const float coef = sf * dinv[f * NN + tv[k]];

#if USE_ASYNC_STAGE
        if (k + 1 < EPW) {                          // stage next edge's row
            async_stage_row(ldsBase + ((k + 1) & 1) * (DD * 4u),
                            (uint32_t)tv[k + 1], ego, lane);
            wait_async1();                          // edge k's row is resident
        } else {
            wait_async0();
        }
        const float4 row = *(const float4*)(&stage[wid * 2 * DD + (k & 1) * DD + lane * 4]);
#else
        const float4 row = *(const float4*)(&ego[(size_t)tv[k] * DD + lane * 4]);
#endif
        float* dst = &x[(size_t)h * DD + lane * 4];
        atomicAdd(dst + 0, row.x * coef);
        atomicAdd(dst + 1, row.y * coef);
        atomicAdd(dst + 2, row.z * coef);
        atomicAdd(dst + 3, row.w * coef);
    }
}

// ---------- fused: x *= d_inv ; rinvx[f][n] = 1/max(||x_chunk||,eps) ----------
__global__ void k_scalenorm(float* __restrict__ x, const float* __restrict__ dinv,
                            float* __restrict__ rinvx) {
    const int lane = threadIdx.x & 31;
    const int n    = blockIdx.x * 8 + (threadIdx.x >> 5);   // grid = NN/8 exactly
#pragma unroll
    for (int f = 0; f < NF; ++f) {
        float dv = dinv[f * NN + n];
        size_t idx = (size_t)n * DD + f * CD + lane;
        float v = x[idx] * dv;
        x[idx] = v;
        float s = wsum32(v * v);
        if (lane == 0) rinvx[f * NN + n] = 1.0f / fmaxf(sqrtf(s), EPSF);
    }
}

// ---------- fallback elementwise scale ----------
__global__ void k_scale(float* __restrict__ x, const float* __restrict__ dinv) {
    size_t i = (size_t)blockIdx.x * blockDim.x + threadIdx.x;   // < NN*DD
    int node = (int)(i >> 7);
    int f    = (int)((i & 127) >> 5);
    x[i] *= dinv[f * NN + node];
}

// ---------- fused attention + next-iteration degree scatter ----------
// A[f][e] += rinvx[f][h] * <x[h]_f, T[t]_f>, then lane0 computes
// softmax(A_new) for this edge and scatters it into deg for the NEXT routing
// iteration — eliminating the standalone k_deg edge pass entirely.
__global__ void k_attn_fused(const float* __restrict__ x, const float* __restrict__ T,
                             const float* __restrict__ rinvx,
                             const int* __restrict__ hl, const int* __restrict__ tl,
                             float* __restrict__ A, float* __restrict__ deg) {
#if USE_ASYNC_STAGE
    __shared__ float sx[8 * DD];
    __shared__ float st[8 * DD];
#endif
    const int lane = threadIdx.x & 31;
    const int wid  = threadIdx.x >> 5;
    const int e    = blockIdx.x * 8 + wid;         // grid = NE/8 exactly
    __builtin_prefetch(&hl[e + 2048], 0, 0);
    __builtin_prefetch(&tl[e + 2048], 0, 0);
    const int h = hl[e];
    const int t = tl[e];
#if USE_ASYNC_STAGE
    async_stage_row((uint32_t)(uintptr_t)(&sx[wid * DD]), (uint32_t)h, x, lane);
    async_stage_row((uint32_t)(uintptr_t)(&st[wid * DD]), (uint32_t)t, T, lane);
    wait_async0();
    const float* xr = &sx[wid * DD];
    const float* tr = &st[wid * DD];
#else
    const float* xr = &x[(size_t)h * DD];
    const float* tr = &T[(size_t)t * DD];
#endif
    float a[NF];
#pragma unroll
    for (int f = 0; f < NF; ++f) {
        float d = wsum32(xr[f * CD + lane] * tr[f * CD + lane]);
        a[f] = A[(size_t)f * NE + e] + d * rinvx[f * NN + h];   // new A value
    }
    if (lane == 0) {
#pragma unroll
        for (int f = 0; f < NF; ++f) A[(size_t)f * NE + e] = a[f];
        float s0, s1, s2, s3;
        softmax4(a[0], a[1], a[2], a[3], s0, s1, s2, s3);
        atomicAdd(&deg[h],          s0);
        atomicAdd(&deg[NN + h],     s1);
        atomicAdd(&deg[2 * NN + h], s2);
        atomicAdd(&deg[3 * NN + h], s3);
    }
}

// ---------- fallback attention (per-edge norms+tanh, unfused) ----------
__global__ void k_attn(const float* __restrict__ x, const float* __restrict__ ego,
                       const int* __restrict__ hl, const int* __restrict__ tl,
                       float* __restrict__ A) {
#if USE_ASYNC_STAGE
    __shared__ float sx[8 * DD];
    __shared__ float se[8 * DD];
#endif
    const int lane = threadIdx.x & 31;
    const int wid  = threadIdx.x >> 5;
    const int e    = blockIdx.x * 8 + wid;         // grid = NE/8 exactly
    const int h = hl[e];
    const int t = tl[e];
#if USE_ASYNC_STAGE
    async_stage_row((uint32_t)(uintptr_t)(&sx[wid * DD]), (uint32_t)h, x,   lane);
    async_stage_row((uint32_t)(uintptr_t)(&se[wid * DD]), (uint32_t)t, ego, lane);
    wait_async0();
    const float* xr = &sx[wid * DD];
    const float* er = &se[wid * DD];
#else
    const float* xr = &x[(size_t)h * DD];
    const float* er = &ego[(size_t)t * DD];
#endif
#pragma unroll
    for (int f = 0; f < NF; ++f) {
        float xv = xr[f * CD + lane];
        float ev = er[f * CD + lane];
        float nx = wsum32(xv * xv);
        float ne = wsum32(ev * ev);
        float hn = xv / fmaxf(sqrtf(nx), EPSF);
        float tn = ev / fmaxf(sqrtf(ne), EPSF);
        float d  = wsum32(hn * tanhf(tn));
        if (lane == 0) A[(size_t)f * NE + e] += d;
    }
}

// ---------- ego = x ; accum += x ----------
__global__ void k_upd(const float* __restrict__ x, float* __restrict__ ego,
                      float* __restrict__ acc) {
    size_t i = (size_t)blockIdx.x * blockDim.x + threadIdx.x;   // < NN*DD
    float v = x[i];
    ego[i] = v;
    acc[i] += v;
}

// ---------- out = accum/3, mirrored into the second (u,i) pair ----------
__global__ void k_final(float* __restrict__ out) {
    size_t i = (size_t)blockIdx.x * blockDim.x + threadIdx.x;   // < NN*DD
    float v = out[i] * (1.0f / 3.0f);
    out[i] = v;
    out[(size_t)NN * DD + i] = v;
}

extern "C" void kernel_launch(void* const* d_in, const int* in_sizes, int n_in,
                              void* d_out, int out_size, void* d_ws, size_t ws_size,
                              hipStream_t stream) {
    const float* ue = (const float*)d_in[0];
    const float* ie = (const float*)d_in[1];
    const int*   hl = (const int*)d_in[2];
    const int*   tl = (const int*)d_in[3];
    float* out = (float*)d_out;

    // Workspace layout: ego | x | A | deg/dinv | rinvx | T
    //   base (fallback) : 136.0 MB ; full (fused path): 188.8 MB (< 192MB L2)
    float* ego   = (float*)d_ws;
    float* x     = ego   + (size_t)NN * DD;
    float* A     = x     + (size_t)NN * DD;
    float* dinv  = A     + (size_t)NF * NE;     // holds deg, transformed in place
    float* rinvx = dinv  + (size_t)NF * NN;
    float* T     = rinvx + (size_t)NF * NN;
    float* acc   = out;                         // accum lives in d_out[0 .. NN*DD)

    const size_t needBig = ((size_t)NN * DD * 3 + (size_t)NF * NE + (size_t)NF * NN * 2)
                           * sizeof(float);
    const bool big = (ws_size >= needBig);

    const int ND_BLOCKS = (NN * DD) / 256;   // 50000 (exact)
    const int A_BLOCKS  = (NF * NE) / 256;   // 31250 (exact)
    const int E_BLOCKS  = (NE + 255) / 256;  // thread/edge
    const int EW_BLOCKS = NE / 8;            // 250000 (exact): wave/edge
    const int EP_BLOCKS = NE / (8 * EPW);    // 62500  (exact): EPW edges/wave
    const int NW_BLOCKS = NN / 8;            // 12500  (exact): wave/node
    const int DV_BLOCKS = (NF * NN + 255) / 256;

    k_init <<<ND_BLOCKS, 256, 0, stream>>>(ue, ie, ego, acc);
    k_fill1<<<A_BLOCKS, 256, 0, stream>>>(A);

    // First-iteration degrees (A==1 -> uniform 0.25 scores, no A read).
    k_zero<<<DV_BLOCKS, 256, 0, stream>>>(dinv, (long)NF * NN);
    k_deg0<<<E_BLOCKS, 256, 0, stream>>>(hl, dinv);

    for (int layer = 0; layer < 2; ++layer) {
        if (big)   // per-layer hoist of tanh(l2norm(ego))
            k_egoprep<<<NW_BLOCKS, 256, 0, stream>>>(ego, T);
        for (int it = 0; it < 2; ++it) {
            const bool last = (layer == 1 && it == 1);
            k_dinv<<<DV_BLOCKS, 256, 0, stream>>>(dinv);            // deg -> d_inv
            k_zero<<<ND_BLOCKS, 256, 0, stream>>>(x, (long)NN * DD);
            k_prop<<<EP_BLOCKS, 256, 0, stream>>>(A, hl, tl, dinv, ego, x);
            if (big)
                k_scalenorm<<<NW_BLOCKS, 256, 0, stream>>>(x, dinv, rinvx);
            else
                k_scale<<<ND_BLOCKS, 256, 0, stream>>>(x, dinv);
            if (!last) {   // final A-update is never consumed
                if (big) {
                    // dinv fully consumed; recycle buffer as next iteration's deg
                    k_zero<<<DV_BLOCKS, 256, 0, stream>>>(dinv, (long)NF * NN);
                    k_attn_fused<<<EW_BLOCKS, 256, 0, stream>>>(x, T, rinvx,
                                                                hl, tl, A, dinv);
                } else {
                    k_attn<<<EW_BLOCKS, 256, 0, stream>>>(x, ego, hl, tl, A);
                    k_zero<<<DV_BLOCKS, 256, 0, stream>>>(dinv, (long)NF * NN);
                    k_deg <<<E_BLOCKS, 256, 0, stream>>>(A, hl, dinv);
                }
            }
        }
        k_upd<<<ND_BLOCKS, 256, 0, stream>>>(x, ego, acc);
    }
    k_final<<<ND_BLOCKS, 256, 0, stream>>>(out);
}